// GCN_11373073400297
// MI455X (gfx1250) — compile-verified
//
#include <hip/hip_runtime.h>
#include <hip/hip_bf16.h>

typedef __attribute__((ext_vector_type(2))) float v2f;
typedef __attribute__((ext_vector_type(8))) float v8f;

// ---------------------------------------------------------------------------
// GCN norm kernels
// ---------------------------------------------------------------------------
__global__ void k_init_deg(float* __restrict__ deg, int N) {
    int i = blockIdx.x * blockDim.x + threadIdx.x;
    if (i < N) deg[i] = 1.0f;                 // self-loop weight
}

__global__ void k_scatter_deg(const int* __restrict__ col,
                              const float* __restrict__ w,
                              float* __restrict__ deg, int E) {
    int e = blockIdx.x * blockDim.x + threadIdx.x;
    if (e < E) atomicAdd(&deg[col[e]], w[e]);
}

__global__ void k_dinv(const float* __restrict__ deg,
                       float* __restrict__ dinv, int N) {
    int i = blockIdx.x * blockDim.x + threadIdx.x;
    if (i < N) {
        float d = deg[i];
        dinv[i] = (d > 0.0f) ? rsqrtf(d) : 0.0f;
    }
}

__global__ void k_norm(const int* __restrict__ row,
                       const int* __restrict__ col,
                       const float* __restrict__ w,
                       const float* __restrict__ dinv,
                       float* __restrict__ norm, int E) {
    int e = blockIdx.x * blockDim.x + threadIdx.x;
    if (e < E) norm[e] = dinv[row[e]] * w[e] * dinv[col[e]];
}

// ---------------------------------------------------------------------------
// GEMM1: h1[N,128] = X[N,128] @ W1[128,128]  (fp32 WMMA 16x16x4)
// Block: 256 threads = 8 waves; block owns 16 rows; wave w owns n-cols [16w,16w+16)
// ---------------------------------------------------------------------------
__global__ __launch_bounds__(256) void k_gemm1(const float* __restrict__ X,
                                               const float* __restrict__ W,
                                               float* __restrict__ H,
                                               int N) {
    __shared__ float Alds[16 * 128];          // 8 KB
    const int tid = threadIdx.x;
    const int bm  = blockIdx.x * 16;

    // stage A tile (16 rows x 128 cols) coalesced, zero-padded at the tail
    float4* al4 = (float4*)Alds;
    for (int i = tid; i < 16 * 32; i += 256) {          // 512 float4
        int r = i >> 5, g = i & 31;
        float4 v = make_float4(0.f, 0.f, 0.f, 0.f);
        if (bm + r < N) v = ((const float4*)(X + (size_t)(bm + r) * 128))[g];
        al4[i] = v;
    }
    __syncthreads();

    const int wave = tid >> 5;
    const int lane = tid & 31;
    const int nb   = wave * 16;
    const int mr   = lane & 15;              // A row / B col / D col index
    const int kh   = (lane >> 4) * 2;        // K sub-offset for upper half-wave

    v8f acc = {};
    for (int kc = 0; kc < 128; kc += 4) {
        v2f a, b;
        a[0] = Alds[mr * 128 + kc + kh + 0];
        a[1] = Alds[mr * 128 + kc + kh + 1];
        b[0] = W[(kc + kh + 0) * 128 + nb + mr];
        b[1] = W[(kc + kh + 1) * 128 + nb + mr];
        acc = __builtin_amdgcn_wmma_f32_16x16x4_f32(
                  false, a, false, b, (short)0, acc, false, false);
    }

    const int mtop = (lane >> 4) * 8;
    for (int v = 0; v < 8; ++v) {
        int m = bm + mtop + v;
        if (m < N) H[(size_t)m * 128 + nb + mr] = acc[v];
    }
}

// ---------------------------------------------------------------------------
// out1[i,f] = b1[f] + dinv[i]^2 * h1[i,f]   (bias + self-loop contribution)
// ---------------------------------------------------------------------------
__global__ void k_init_out1(const float* __restrict__ h1,
                            const float* __restrict__ dinv,
                            const float* __restrict__ b1,
                            float* __restrict__ out1, int N) {
    int idx = blockIdx.x * blockDim.x + threadIdx.x;   // over N * 32 float4-groups
    if (idx >= N * 32) return;
    int i = idx >> 5, g = idx & 31;
    float s = dinv[i]; s *= s;
    float4 h = ((const float4*)(h1 + (size_t)i * 128))[g];
    float4 b = ((const float4*)b1)[g];
    float4 o = make_float4(b.x + h.x * s, b.y + h.y * s,
                           b.z + h.z * s, b.w + h.w * s);
    ((float4*)(out1 + (size_t)i * 128))[g] = o;
}

// edge scatter, layer 1: out1[col] += norm * h1[row], F=128 (32 float4 groups)
__global__ void k_scatter1(const float* __restrict__ h1,
                           const int* __restrict__ row,
                           const int* __restrict__ col,
                           const float* __restrict__ norm,
                           float* __restrict__ out1, int E) {
    int idx = blockIdx.x * blockDim.x + threadIdx.x;   // over E * 32
    if (idx >= E * 32) return;
    int e = idx >> 5, g = idx & 31;
    int r = row[e], c = col[e];
    float nw = norm[e];
    float4 v = ((const float4*)(h1 + (size_t)r * 128))[g];
    float* o = out1 + (size_t)c * 128 + g * 4;
    atomicAdd(o + 0, v.x * nw);
    atomicAdd(o + 1, v.y * nw);
    atomicAdd(o + 2, v.z * nw);
    atomicAdd(o + 3, v.w * nw);
}

// ---------------------------------------------------------------------------
// GEMM2: h2[N,64] = relu(out1[N,128]) @ W2[128,64]
// Block: 256 threads = 8 waves; block owns 32 rows; wave w -> m-sub (w/4)*16, n (w%4)*16
// ---------------------------------------------------------------------------
__global__ __launch_bounds__(256) void k_gemm2(const float* __restrict__ A,
                                               const float* __restrict__ W2,
                                               float* __restrict__ H2,
                                               int N) {
    __shared__ float Alds[32 * 128];          // 16 KB
    __shared__ float Wlds[128 * 64];          // 32 KB
    const int tid = threadIdx.x;
    const int bm  = blockIdx.x * 32;

    // stage W2 (8192 floats)
    {
        const float4* w4 = (const float4*)W2;
        float4* wl = (float4*)Wlds;
        for (int i = tid; i < 2048; i += 256) wl[i] = w4[i];
    }
    // stage A with fused ReLU, zero-padded
    {
        float4* al = (float4*)Alds;
        for (int i = tid; i < 32 * 32; i += 256) {     // 1024 float4
            int r = i >> 5, g = i & 31;
            float4 v = make_float4(0.f, 0.f, 0.f, 0.f);
            if (bm + r < N) v = ((const float4*)(A + (size_t)(bm + r) * 128))[g];
            v.x = fmaxf(v.x, 0.f); v.y = fmaxf(v.y, 0.f);
            v.z = fmaxf(v.z, 0.f); v.w = fmaxf(v.w, 0.f);
            al[i] = v;
        }
    }
    __syncthreads();

    const int wave = tid >> 5;
    const int lane = tid & 31;
    const int msub = (wave >> 2) * 16;
    const int nb   = (wave & 3) * 16;
    const int mr   = lane & 15;
    const int kh   = (lane >> 4) * 2;

    v8f acc = {};
    for (int kc = 0; kc < 128; kc += 4) {
        v2f a, b;
        a[0] = Alds[(msub + mr) * 128 + kc + kh + 0];
        a[1] = Alds[(msub + mr) * 128 + kc + kh + 1];
        b[0] = Wlds[(kc + kh + 0) * 64 + nb + mr];
        b[1] = Wlds[(kc + kh + 1) * 64 + nb + mr];
        acc = __builtin_amdgcn_wmma_f32_16x16x4_f32(
                  false, a, false, b, (short)0, acc, false, false);
    }

    const int mtop = (lane >> 4) * 8;
    for (int v = 0; v < 8; ++v) {
        int m = bm + msub + mtop + v;
        if (m < N) H2[(size_t)m * 64 + nb + mr] = acc[v];
    }
}

// out[i,f] = b2[f] + dinv[i]^2 * h2[i,f]   (F = 64 -> 16 float4 groups)
__global__ void k_init_out2(const float* __restrict__ h2,
                            const float* __restrict__ dinv,
                            const float* __restrict__ b2,
                            float* __restrict__ out, int N) {
    int idx = blockIdx.x * blockDim.x + threadIdx.x;   // over N * 16
    if (idx >= N * 16) return;
    int i = idx >> 4, g = idx & 15;
    float s = dinv[i]; s *= s;
    float4 h = ((const float4*)(h2 + (size_t)i * 64))[g];
    float4 b = ((const float4*)b2)[g];
    float4 o = make_float4(b.x + h.x * s, b.y + h.y * s,
                           b.z + h.z * s, b.w + h.w * s);
    ((float4*)(out + (size_t)i * 64))[g] = o;
}

// edge scatter, layer 2: out[col] += norm * h2[row], F=64 (16 float4 groups)
__global__ void k_scatter2(const float* __restrict__ h2,
                           const int* __restrict__ row,
                           const int* __restrict__ col,
                           const float* __restrict__ norm,
                           float* __restrict__ out, int E) {
    int idx = blockIdx.x * blockDim.x + threadIdx.x;   // over E * 16
    if (idx >= E * 16) return;
    int e = idx >> 4, g = idx & 15;
    int r = row[e], c = col[e];
    float nw = norm[e];
    float4 v = ((const float4*)(h2 + (size_t)r * 64))[g];
    float* o = out + (size_t)c * 64 + g * 4;
    atomicAdd(o + 0, v.x * nw);
    atomicAdd(o + 1, v.y * nw);
    atomicAdd(o + 2, v.z * nw);
    atomicAdd(o + 3, v.w * nw);
}

// ---------------------------------------------------------------------------
// Launch
// ---------------------------------------------------------------------------
extern "C" void kernel_launch(void* const* d_in, const int* in_sizes, int n_in,
                              void* d_out, int out_size, void* d_ws, size_t ws_size,
                              hipStream_t stream) {
    const float* x   = (const float*)d_in[0];
    const int*   ei  = (const int*)d_in[1];     // [2, E] flat
    const float* ew  = (const float*)d_in[2];
    const float* W1  = (const float*)d_in[3];
    const float* b1  = (const float*)d_in[4];
    const float* W2  = (const float*)d_in[5];
    const float* b2  = (const float*)d_in[6];
    float* out = (float*)d_out;

    const int E = in_sizes[2];                  // 800000
    const int N = in_sizes[0] / 128;            // 50000
    const int* row = ei;                        // edge_index[0]
    const int* col = ei + E;                    // edge_index[1]

    float* ws   = (float*)d_ws;
    float* deg  = ws;                           // N
    float* dinv = ws + N;                       // N
    float* nrm  = ws + 2 * (size_t)N;           // E
    float* h1   = nrm + E;                      // N*128
    float* out1 = h1 + (size_t)N * 128;         // N*128
    float* h2   = out1 + (size_t)N * 128;       // N*64

    const int T = 256;

    // --- normalization coefficients ---
    k_init_deg<<<(N + T - 1) / T, T, 0, stream>>>(deg, N);
    k_scatter_deg<<<(E + T - 1) / T, T, 0, stream>>>(col, ew, deg, E);
    k_dinv<<<(N + T - 1) / T, T, 0, stream>>>(deg, dinv, N);
    k_norm<<<(E + T - 1) / T, T, 0, stream>>>(row, col, ew, dinv, nrm, E);

    // --- layer 1 ---
    k_gemm1<<<(N + 15) / 16, 256, 0, stream>>>(x, W1, h1, N);
    k_init_out1<<<((size_t)N * 32 + T - 1) / T, T, 0, stream>>>(h1, dinv, b1, out1, N);
    k_scatter1<<<((size_t)E * 32 + T - 1) / T, T, 0, stream>>>(h1, row, col, nrm, out1, E);

    // --- layer 2 (ReLU fused into GEMM2 A-load) ---
    k_gemm2<<<(N + 31) / 32, 256, 0, stream>>>(out1, W2, h2, N);
    k_init_out2<<<((size_t)N * 16 + T - 1) / T, T, 0, stream>>>(h2, dinv, b2, out, N);
    k_scatter2<<<((size_t)E * 16 + T - 1) / T, T, 0, stream>>>(h2, row, col, nrm, out, E);
}